// Image_Tokenizer_87213605912837
// MI455X (gfx1250) — compile-verified
//
#include <hip/hip_runtime.h>

#define N_TOK 8192
#define DIM   4096
#define VOCAB 8192
#define DECAY 0.99f
#define EPS   1e-5f
#define CCOST 0.25f

#define TILE_M 32
#define TILE_N 256
#define KT     32
#define LSTR   36   // LDS row stride in floats (32 data + 4 pad) == TDM pad 32dw -> +4dw

typedef float v2f   __attribute__((ext_vector_type(2)));
typedef float v8f   __attribute__((ext_vector_type(8)));
typedef float f32x4 __attribute__((ext_vector_type(4)));

typedef unsigned int tdm_u32x4 __attribute__((ext_vector_type(4)));
typedef int          tdm_i32x8 __attribute__((ext_vector_type(8)));
typedef int          tdm_i32x4 __attribute__((ext_vector_type(4)));

// ---------------- TDM issue: DMA a 256x32 f32 tile of E into LDS ----------------
// pad_interval = 32 DWORDs, pad_amount = 4 DWORDs  ->  LDS row stride = 36 floats (LSTR)
__device__ __forceinline__ void tdm_issue_e_tile(const float* E, int n0, int k0,
                                                 unsigned lds_addr) {
  const unsigned long long ga =
      (unsigned long long)(uintptr_t)(E + (size_t)n0 * DIM + k0);
  tdm_u32x4 g0;
  g0[0] = 1u;                                        // count=1 valid descriptor
  g0[1] = lds_addr;                                  // lds_addr (bytes)
  g0[2] = (unsigned)(ga & 0xFFFFFFFFu);              // global_addr[31:0]
  g0[3] = (unsigned)((ga >> 32) & 0x01FFFFFFu)       // global_addr[56:32]
        | (2u << 30);                                // type = 2 (image)
  tdm_i32x8 g1;
  g1[0] = (int)((2u << 16)      // data_size = 4B
              | (1u << 20)      // pad_enable
              | (4u << 22)      // pad_interval: 32 DWORDs
              | (3u << 25));    // pad_amount:   4 DWORDs
  g1[1] = (int)(((unsigned)DIM & 0xFFFFu) << 16);    // tensor_dim0[15:0]
  g1[2] = (int)((((unsigned)DIM >> 16) & 0xFFFFu)    // tensor_dim0[31:16]
              | (((unsigned)VOCAB & 0xFFFFu) << 16));// tensor_dim1[15:0]
  g1[3] = (int)((((unsigned)VOCAB >> 16) & 0xFFFFu)  // tensor_dim1[31:16]
              | ((unsigned)KT << 16));               // tile_dim0 = 32
  g1[4] = (int)(unsigned)TILE_N;                     // tile_dim1 = 256, tile_dim2 = 0
  g1[5] = (int)(unsigned)DIM;                        // tensor_dim0_stride[31:0]
  g1[6] = 0;
  g1[7] = 0;
  const tdm_i32x4 zz4 = {0, 0, 0, 0};
#if defined(__clang_major__) && (__clang_major__ >= 23)
  const tdm_i32x8 zz8 = {0, 0, 0, 0, 0, 0, 0, 0};
  __builtin_amdgcn_tensor_load_to_lds(g0, g1, zz4, zz4, zz8, 0);
#else
  __builtin_amdgcn_tensor_load_to_lds(g0, g1, zz4, zz4, 0);
#endif
}

// ---------------- block reduction (deterministic, blockDim == 256) ----------------
__device__ __forceinline__ float blockReduceSum(float v) {
  __shared__ float red[8];
  const int lane = threadIdx.x & 31;
  const int w    = threadIdx.x >> 5;
  #pragma unroll
  for (int off = 16; off; off >>= 1) v += __shfl_xor(v, off, 32);
  __syncthreads();
  if (lane == 0) red[w] = v;
  __syncthreads();
  float r = 0.0f;
  if (threadIdx.x == 0) {
    #pragma unroll
    for (int i = 0; i < 8; ++i) r += red[i];
  }
  __syncthreads();
  return r;  // valid in thread 0
}

// ---------------- row norms: ||x_i||^2 and ||e_j||^2 ----------------
__global__ __launch_bounds__(256) void vq_norms(const float* __restrict__ X,
                                                const float* __restrict__ E,
                                                float* __restrict__ x2,
                                                float* __restrict__ e2) {
  const int row = blockIdx.x;
  const float* p = (row < N_TOK) ? (X + (size_t)row * DIM)
                                 : (E + (size_t)(row - N_TOK) * DIM);
  float s = 0.0f;
  for (int d = threadIdx.x; d < DIM; d += 256) { float v = p[d]; s += v * v; }
  float r = blockReduceSum(s);
  if (threadIdx.x == 0) {
    if (row < N_TOK) x2[row] = r; else e2[row - N_TOK] = r;
  }
}

// ------- fused distance GEMM (fp32 WMMA, double-buffered TDM staging) + argmin -------
__global__ __launch_bounds__(256) void vq_gemm_argmin(
    const float* __restrict__ X, const float* __restrict__ E,
    const float* __restrict__ x2, const float* __restrict__ e2,
    int* __restrict__ out_idx) {
  __shared__ float xs[2][TILE_M * LSTR];
  __shared__ float es[2][TILE_N * LSTR];
  __shared__ float rv[TILE_M][4];
  __shared__ int   ri[TILE_M][4];

  const int tid   = threadIdx.x;
  const int lane  = tid & 31;
  const int wave  = tid >> 5;    // 0..7
  const int wr    = wave >> 2;   // 0..1 : which 16-row slab
  const int wc    = wave & 3;    // 0..3 : which 64-col slab
  const int lhalf = lane >> 4;   // 0..1
  const int l16   = lane & 15;
  const int m0    = blockIdx.x * TILE_M;

  // LDS byte offsets of the two E staging buffers (flat LDS address truncates to offset)
  const unsigned es_lds[2] = {(unsigned)(uintptr_t)(void*)&es[0][0],
                              (unsigned)(uintptr_t)(void*)&es[1][0]};

  // rows touched by this lane's accumulator elements (C/D layout: M = v + 8*(lane/16))
  float x2row[8];
  #pragma unroll
  for (int v = 0; v < 8; ++v) x2row[v] = x2[m0 + 16 * wr + 8 * lhalf + v];

  float minv[8];
  int   mini[8];
  #pragma unroll
  for (int v = 0; v < 8; ++v) { minv[v] = 3.402823466e38f; mini[v] = 0; }

  const int ldr = tid >> 3;         // 0..31
  const int ldc = (tid & 7) << 2;   // 0,4,...,28

  // ---- prologue: start stage (n0=0, k0=0) into buffer 0 ----
  if (wave == 0) tdm_issue_e_tile(E, 0, 0, es_lds[0]);
  {
    f32x4 d = *(const f32x4*)(X + (size_t)(m0 + ldr) * DIM + 0 + ldc);
    *(f32x4*)(&xs[0][0] + ldr * LSTR + ldc) = d;
  }

  int buf = 0;
  for (int n0 = 0; n0 < VOCAB; n0 += TILE_N) {
    const v8f z = {0.f, 0.f, 0.f, 0.f, 0.f, 0.f, 0.f, 0.f};
    v8f acc[4] = {z, z, z, z};

    for (int k0 = 0; k0 < DIM; k0 += KT) {
      // coordinates of the next stage (rolls across the column-tile boundary)
      int nn = n0, nk = k0 + KT;
      if (nk == DIM) { nk = 0; nn = n0 + TILE_N; }
      const bool has_next = (nn < VOCAB);

      __syncthreads();   // all waves finished computing on buf^1 -> safe to overwrite

      if (has_next) {
        if (wave == 0) tdm_issue_e_tile(E, nn, nk, es_lds[buf ^ 1]);
        f32x4 d = *(const f32x4*)(X + (size_t)(m0 + ldr) * DIM + nk + ldc);
        *(f32x4*)(&xs[buf ^ 1][0] + ldr * LSTR + ldc) = d;
        // speculative prefetch two stages ahead of X
        __builtin_prefetch(X + (size_t)(m0 + ldr) * DIM + ((nk + KT) & (DIM - 1)) + ldc, 0, 1);
      }
      // TDM completes in order: <=1 outstanding means *current* buffer is resident,
      // while the next stage keeps streaming in the background during the WMMAs.
      if (wave == 0) {
        if (has_next) __builtin_amdgcn_s_wait_tensorcnt(1);
        else          __builtin_amdgcn_s_wait_tensorcnt(0);
      }
      __syncthreads();   // current buffer visible to all waves

      // A(16x4 f32): lane l, vgpr v -> M = l%16, K = 2*(l/16)+v  (2 consecutive f32 -> b64)
      // B(4x16) = E^T, E row-major -> same lane->(row,k) map as A.
      const float* xsb = &xs[buf][0];
      const float* esb = &es[buf][0];
      #pragma unroll
      for (int kk = 0; kk < KT; kk += 4) {
        const int kl = kk + 2 * lhalf;
        v2f a = *(const v2f*)(xsb + (16 * wr + l16) * LSTR + kl);
        #pragma unroll
        for (int t = 0; t < 4; ++t) {
          v2f b = *(const v2f*)(esb + (64 * wc + 16 * t + l16) * LSTR + kl);
          acc[t] = __builtin_amdgcn_wmma_f32_16x16x4_f32(
              false, a, false, b, (short)0, acc[t], false, false);
        }
      }
      buf ^= 1;
    }

    // fold distances into running per-lane argmin
    #pragma unroll
    for (int t = 0; t < 4; ++t) {
      const int col = n0 + 64 * wc + 16 * t + l16;
      const float e2c = e2[col];
      #pragma unroll
      for (int v = 0; v < 8; ++v) {
        const float d = x2row[v] + e2c - 2.0f * acc[t][v];
        if (d < minv[v]) { minv[v] = d; mini[v] = col; }
      }
    }
  }

  // reduce argmin across the 16 lanes sharing a row (tie -> lowest index)
  #pragma unroll
  for (int v = 0; v < 8; ++v) {
    float bv = minv[v]; int bi = mini[v];
    #pragma unroll
    for (int off = 8; off; off >>= 1) {
      const float ov = __shfl_xor(bv, off, 32);
      const int   oi = __shfl_xor(bi, off, 32);
      if (ov < bv || (ov == bv && oi < bi)) { bv = ov; bi = oi; }
    }
    if (l16 == 0) {
      const int r = 16 * wr + 8 * lhalf + v;
      rv[r][wc] = bv; ri[r][wc] = bi;
    }
  }
  __syncthreads();
  // reduce across the 4 column-slab waves
  if (tid < TILE_M) {
    float bv = rv[tid][0]; int bi = ri[tid][0];
    #pragma unroll
    for (int c = 1; c < 4; ++c) {
      const float ov = rv[tid][c]; const int oi = ri[tid][c];
      if (ov < bv || (ov == bv && oi < bi)) { bv = ov; bi = oi; }
    }
    out_idx[m0 + tid] = bi;
  }
}

// ---------------- counts ----------------
__global__ void vq_zero_cnt(int* __restrict__ cnt) {
  cnt[blockIdx.x * blockDim.x + threadIdx.x] = 0;
}

__global__ void vq_hist(const int* __restrict__ idx, int* __restrict__ cnt) {
  const int i = blockIdx.x * blockDim.x + threadIdx.x;
  atomicAdd(&cnt[idx[i]], 1);  // integer atomics: deterministic
}

// ---------------- EMA cluster size (raw) + partial sums for n and entropy ----------------
__global__ __launch_bounds__(256) void vq_cs_raw(const float* __restrict__ ecs,
                                                 const int* __restrict__ cnt,
                                                 float* __restrict__ raw,
                                                 float* __restrict__ pn,
                                                 float* __restrict__ pe) {
  float sn = 0.f, se = 0.f;
  for (int j = blockIdx.x * 256 + threadIdx.x; j < VOCAB; j += gridDim.x * 256) {
    const float c = (float)cnt[j];
    const float r = ecs[j] * DECAY + (1.0f - DECAY) * c;
    raw[j] = r;
    sn += r;
    const float p = c * (1.0f / (float)N_TOK);
    se += p * logf(p + 1e-10f);
  }
  const float rn = blockReduceSum(sn);
  const float re = blockReduceSum(se);
  if (threadIdx.x == 0) { pn[blockIdx.x] = rn; pe[blockIdx.x] = re; }
}

__global__ void vq_cs_fin(const float* __restrict__ pn, const float* __restrict__ pe,
                          float* __restrict__ nscal, float* __restrict__ out_perp) {
  if (threadIdx.x == 0) {
    float n = 0.f, e = 0.f;
    for (int i = 0; i < 32; ++i) { n += pn[i]; e += pe[i]; }
    nscal[0] = n;
    out_perp[0] = expf(-e);
  }
}

__global__ void vq_cs_norm(const float* __restrict__ raw, const float* __restrict__ nscal,
                           float* __restrict__ out_ncs) {
  const int j = blockIdx.x * blockDim.x + threadIdx.x;
  const float n = nscal[0];
  out_ncs[j] = (raw[j] + EPS) / (n + (float)VOCAB * EPS) * n;
}

// ---------------- dw scatter (atomic-free, per-code scan) + new_ema_w + new_embedding ----------------
__global__ __launch_bounds__(256) void vq_ema(const float* __restrict__ X,
                                              const float* __restrict__ ema_w,
                                              const int* __restrict__ idx,
                                              const float* __restrict__ ncs,
                                              float* __restrict__ out_emaw,
                                              float* __restrict__ out_emb) {
  const int j   = blockIdx.x;
  const int tid = threadIdx.x;
  __shared__ int sidx[256];
  float acc[16];
  #pragma unroll
  for (int u = 0; u < 16; ++u) acc[u] = 0.f;

  for (int base = 0; base < N_TOK; base += 256) {
    __syncthreads();
    sidx[tid] = idx[base + tid];
    const int any = __syncthreads_count(sidx[tid] == j);
    if (any) {
      for (int c = 0; c < 256; ++c) {
        if (sidx[c] == j) {
          const float* xr = X + (size_t)(base + c) * DIM;
          #pragma unroll
          for (int u = 0; u < 16; ++u) acc[u] += xr[tid + (u << 8)];
        }
      }
    }
  }

  const float inv = 1.0f / ncs[j];
  const float* wrow = ema_w + (size_t)j * DIM;
  float* oa = out_emaw + (size_t)j * DIM;
  float* ob = out_emb  + (size_t)j * DIM;
  #pragma unroll
  for (int u = 0; u < 16; ++u) {
    const int d = tid + (u << 8);
    const float nw = wrow[d] * DECAY + (1.0f - DECAY) * acc[u];
    oa[d] = nw;
    ob[d] = nw * inv;
  }
}

// ---------------- quantized_st gather + loss partials + float indices ----------------
__global__ __launch_bounds__(256) void vq_qst(const float* __restrict__ X,
                                              const float* __restrict__ E,
                                              const int* __restrict__ idx,
                                              float* __restrict__ out_q,
                                              float* __restrict__ out_idxf,
                                              float* __restrict__ lp) {
  const int i   = blockIdx.x;
  const int j   = idx[i];
  const int tid = threadIdx.x;
  const float* er = E + (size_t)j * DIM;
  const float* xr = X + (size_t)i * DIM;
  float* qr = out_q + (size_t)i * DIM;
  float s = 0.f;
  #pragma unroll
  for (int u = 0; u < 16; ++u) {
    const int d = tid + (u << 8);
    const float q = er[d];
    const float x = xr[d];
    qr[d] = q;
    const float df = q - x;
    s += df * df;
  }
  const float r = blockReduceSum(s);
  if (tid == 0) { lp[i] = r; out_idxf[i] = (float)j; }
}

__global__ __launch_bounds__(256) void vq_loss_fin(const float* __restrict__ lp,
                                                   float* __restrict__ out_loss) {
  float s = 0.f;
  for (int i = threadIdx.x; i < N_TOK; i += 256) s += lp[i];
  const float r = blockReduceSum(s);
  if (threadIdx.x == 0)
    out_loss[0] = CCOST * r / ((float)N_TOK * (float)DIM);
}

// ---------------- launcher ----------------
extern "C" void kernel_launch(void* const* d_in, const int* in_sizes, int n_in,
                              void* d_out, int out_size, void* d_ws, size_t ws_size,
                              hipStream_t stream) {
  (void)in_sizes; (void)n_in; (void)out_size; (void)ws_size;
  const float* X   = (const float*)d_in[0];  // inputs        [N, D]
  const float* E   = (const float*)d_in[1];  // embedding     [V, D]
  const float* EW  = (const float*)d_in[2];  // ema_w         [V, D]
  const float* ECS = (const float*)d_in[3];  // ema_cluster   [V]
  float* out = (float*)d_out;

  const size_t NTD = (size_t)N_TOK * DIM;
  const size_t VD  = (size_t)VOCAB * DIM;
  float* o_loss = out;                 // 1
  float* o_qst  = out + 1;             // N*D
  float* o_perp = out + 1 + NTD;       // 1
  float* o_idx  = out + 2 + NTD;       // N
  float* o_ncs  = o_idx + N_TOK;       // V
  float* o_emaw = o_ncs + VOCAB;       // V*D
  float* o_emb  = o_emaw + VD;         // V*D

  float* w      = (float*)d_ws;
  float* ws_x2  = w;                   // 8192
  float* ws_e2  = w + 8192;            // 8192
  int*   ws_idx = (int*)(w + 16384);   // 8192
  int*   ws_cnt = (int*)(w + 24576);   // 8192
  float* ws_raw = w + 32768;           // 8192
  float* ws_pn  = w + 40960;           // 32
  float* ws_pe  = w + 40992;           // 32
  float* ws_n   = w + 41024;           // 1
  float* ws_lp  = w + 41088;           // 8192

  vq_zero_cnt   <<<VOCAB / 256, 256, 0, stream>>>(ws_cnt);
  vq_norms      <<<N_TOK + VOCAB, 256, 0, stream>>>(X, E, ws_x2, ws_e2);
  vq_gemm_argmin<<<N_TOK / TILE_M, 256, 0, stream>>>(X, E, ws_x2, ws_e2, ws_idx);
  vq_hist       <<<N_TOK / 256, 256, 0, stream>>>(ws_idx, ws_cnt);
  vq_cs_raw     <<<32, 256, 0, stream>>>(ECS, ws_cnt, ws_raw, ws_pn, ws_pe);
  vq_cs_fin     <<<1, 32, 0, stream>>>(ws_pn, ws_pe, ws_n, o_perp);
  vq_cs_norm    <<<VOCAB / 256, 256, 0, stream>>>(ws_raw, ws_n, o_ncs);
  vq_ema        <<<VOCAB, 256, 0, stream>>>(X, EW, ws_idx, o_ncs, o_emaw, o_emb);
  vq_qst        <<<N_TOK, 256, 0, stream>>>(X, E, ws_idx, o_qst, o_idx, ws_lp);
  vq_loss_fin   <<<1, 256, 0, stream>>>(ws_lp, o_loss);
}